// PRCN_layer_20014547599944
// MI455X (gfx1250) — compile-verified
//
#include <hip/hip_runtime.h>

typedef __attribute__((ext_vector_type(2))) float v2f;
typedef __attribute__((ext_vector_type(8))) float v8f;

#define N_CH     64
#define OUT_CH   64
#define CONV_CH  4096
#define EXPANSION 8192
#define GSZ      8
#define AVG      16
#define HH       56
#define WW       56
#define HW       (HH * WW)       // 3136
#define PIX      16              // pixels per block (N dim of WMMA)
#define CSTRIDE  18              // LDS row stride (floats): 8B-aligned pixel pairs,
                                 // store banks disjoint (+16 mod 64), gather conflicts ~1/32
#define NTILES   (HW / PIX)      // 196

// Dynamic LDS layout:
//   float conv[CONV_CH][CSTRIDE]   : 4096*18*4 = 294912 B
//   int   sidx[EXPANSION]          : 32768 B
// total 327680 B == 320 KB (CDNA5 WGP LDS limit, exact fit)

__global__ __launch_bounds__(256)
void prcn_fused_kernel(const float* __restrict__ x,
                       const float* __restrict__ Wf,
                       const float* __restrict__ bias,
                       const int*   __restrict__ index,
                       float*       __restrict__ out)
{
    extern __shared__ float smem[];
    int* sidx = (int*)(smem + CONV_CH * CSTRIDE);

    const int bb   = blockIdx.x / NTILES;   // batch
    const int tile = blockIdx.x % NTILES;   // 16-pixel tile of flattened H*W
    const int pix0 = tile * PIX;

    const int tid  = threadIdx.x;
    const int lane = tid & 31;
    const int wv   = tid >> 5;      // wave 0..7
    const int n    = lane & 15;     // matrix column = pixel within tile
    const int hi   = lane >> 4;     // half-wave selector (K split per ISA layout)

    // Stage gather indices into LDS (reused 128x per output value in phase 2).
    for (int i = tid; i < EXPANSION; i += 256) sidx[i] = index[i];

    // Pixel coordinates for this lane's column.
    const int p  = pix0 + n;
    const int ph = p / WW;
    const int pw = p % WW;

    // Per-lane im2col offsets + validity masks, computed ONCE (invariant over ic).
    // Tap t = ks*4 + hi*2 + j ; taps 9..11 are K-padding. Invalid taps load offset 0
    // (always legal) and are zeroed via v_cndmask -> no exec divergence.
    int  xoff[3][2];
    bool xok [3][2];
    int  woff[3][2];
    bool wok [3][2];
    #pragma unroll
    for (int ks = 0; ks < 3; ++ks) {
        #pragma unroll
        for (int j = 0; j < 2; ++j) {
            const int t = ks * 4 + hi * 2 + j;
            const bool tv = (t < 9);
            const int tc = tv ? t : 0;
            const int dh = tc / 3 - 1, dw = tc % 3 - 1;
            const int h2 = ph + dh, w2 = pw + dw;
            const bool inb = tv && (unsigned)h2 < (unsigned)HH && (unsigned)w2 < (unsigned)WW;
            xoff[ks][j] = inb ? (h2 * WW + w2) : 0;
            xok [ks][j] = inb;
            woff[ks][j] = tc;
            wok [ks][j] = tv;
        }
    }

    // ---------------- Phase 1: grouped 3x3 conv via V_WMMA_F32_16X16X4_F32 ----------------
    // Per input channel ic: D[64x16] = W[64x9] * im2col[9x16], K padded 9->12 (3 steps of 4).
    // Wave `wv` owns input channels [wv*8, wv*8+8). Unroll x2 so the next channel's
    // global loads overlap the current channel's WMMA chain.
    #pragma unroll 2
    for (int ici = 0; ici < 8; ++ici) {
        const int ic = wv * 8 + ici;
        const float* xc = x + (size_t)(bb * N_CH + ic) * HW;

        // --- Issue ALL global loads for this ic up front, into distinct regs -------------
        // B taps (6 loads):
        float xv[3][2];
        #pragma unroll
        for (int ks = 0; ks < 3; ++ks)
            #pragma unroll
            for (int j = 0; j < 2; ++j)
                xv[ks][j] = xc[xoff[ks][j]];

        // A fragments for all 4 M-tiles (12 v2f, 24 loads/lane, 20 distinct, one clause):
        float wvv[4][3][2];
        #pragma unroll
        for (int mt = 0; mt < 4; ++mt) {
            const int chA = ic * 64 + mt * 16 + (lane & 15);   // A row = filter
            #pragma unroll
            for (int ks = 0; ks < 3; ++ks)
                #pragma unroll
                for (int j = 0; j < 2; ++j)
                    wvv[mt][ks][j] = Wf[chA * 9 + woff[ks][j]];
        }

        // --- Mask-select into fragments (VALU only, no memory waits after this) ----------
        v2f bfr[3];
        #pragma unroll
        for (int ks = 0; ks < 3; ++ks)
            #pragma unroll
            for (int j = 0; j < 2; ++j)
                bfr[ks][j] = xok[ks][j] ? xv[ks][j] : 0.0f;

        v2f afr[4][3];
        #pragma unroll
        for (int mt = 0; mt < 4; ++mt)
            #pragma unroll
            for (int ks = 0; ks < 3; ++ks)
                #pragma unroll
                for (int j = 0; j < 2; ++j)
                    afr[mt][ks][j] = wok[ks][j] ? wvv[mt][ks][j] : 0.0f;

        // --- 12 back-to-back WMMAs, then bias + LDS park per tile ------------------------
        #pragma unroll
        for (int mt = 0; mt < 4; ++mt) {
            v8f acc = {};
            #pragma unroll
            for (int ks = 0; ks < 3; ++ks)
                acc = __builtin_amdgcn_wmma_f32_16x16x4_f32(
                        false, afr[mt][ks], false, bfr[ks], (short)0, acc, false, false);

            // D layout: VGPR j holds row M = j + 8*hi, column = n. Add bias, park in LDS.
            #pragma unroll
            for (int j = 0; j < 8; ++j) {
                const int ch = ic * 64 + mt * 16 + hi * 8 + j;
                smem[ch * CSTRIDE + n] = acc[j] + bias[ch];
            }
        }
    }

    __syncthreads();

    // ---------------- Phase 2: gather -> max(G=8) -> mean(16), all in LDS, b64 loads ------
    // Thread t: pixel pair pp = t&7 (pixels 2pp, 2pp+1), oc group ocb = t>>3 (0..31),
    // handles oc = ocb*2 + q for q in {0,1}. ds_load_b64 halves LDS issue count.
    const int pp  = tid & 7;
    const int ocb = tid >> 3;   // 0..31
    const float* __restrict__ srow = smem + 2 * pp;
    #pragma unroll
    for (int q = 0; q < 2; ++q) {
        const int oc = ocb * 2 + q;
        v2f sum = {0.0f, 0.0f};
        #pragma unroll 4
        for (int a = 0; a < AVG; ++a) {
            v2f mx = {-3.402823466e+38f, -3.402823466e+38f};
            #pragma unroll
            for (int g = 0; g < GSZ; ++g) {
                const int e  = sidx[(oc * AVG + a) * GSZ + g];
                const int ch = e & (CONV_CH - 1);   // e % 4096 (e is a non-negative perm value)
                const v2f vv = *(const v2f*)(srow + ch * CSTRIDE);  // 8B-aligned ds_load_b64
                mx[0] = fmaxf(mx[0], vv[0]);
                mx[1] = fmaxf(mx[1], vv[1]);
            }
            sum[0] += mx[0];
            sum[1] += mx[1];
        }
        v2f r;
        r[0] = sum[0] * (1.0f / AVG);
        r[1] = sum[1] * (1.0f / AVG);
        *(v2f*)(out + (size_t)(bb * OUT_CH + oc) * HW + pix0 + 2 * pp) = r;  // b64 store
    }
}

extern "C" void kernel_launch(void* const* d_in, const int* in_sizes, int n_in,
                              void* d_out, int out_size, void* d_ws, size_t ws_size,
                              hipStream_t stream) {
    const float* x     = (const float*)d_in[0];
    const float* Wf    = (const float*)d_in[1];
    const float* bias  = (const float*)d_in[2];
    const int*   index = (const int*)d_in[3];
    float*       out   = (float*)d_out;

    const size_t shmem = (size_t)CONV_CH * CSTRIDE * sizeof(float)
                       + (size_t)EXPANSION * sizeof(int);     // 327680 B == 320 KB

    // Opt in to the full 320 KB CDNA5 WGP LDS.
    (void)hipFuncSetAttribute((const void*)prcn_fused_kernel,
                              hipFuncAttributeMaxDynamicSharedMemorySize,
                              (int)shmem);

    dim3 grid(8 /*batch*/ * NTILES);
    prcn_fused_kernel<<<grid, 256, shmem, stream>>>(x, Wf, bias, index, out);
}